// Net2_128849019558
// MI455X (gfx1250) — compile-verified
//
#include <hip/hip_runtime.h>
#include <hip/hip_bf16.h>

// GCN (3x GCNConv(128,128) + Linear(128,1)) for MI455X / gfx1250.
//
// Roofline: GEMMs are 4.9 GFLOP fp32 total (negligible on the WMMA pipe);
// the edge scatter moves ~2.6 GB but the whole working set (~58 MB) lives in
// the 192 MB L2, so L2 atomic throughput bounds us. Therefore: exact fp32
// everywhere, V_WMMA_F32_16X16X4_F32 for the dense GEMMs, feature-parallel
// global_atomic_add_f32 for the scatter, self-loop folded into the epilogue.

typedef float v2f __attribute__((ext_vector_type(2)));
typedef float v8f __attribute__((ext_vector_type(8)));

#define HD 128

// ---------------- degree / normalization ----------------

__global__ void k_zero_f32(float* __restrict__ p, int n) {
  int i = blockIdx.x * blockDim.x + threadIdx.x;
  if (i < n) p[i] = 0.0f;
}

__global__ void k_deg_count(const int* __restrict__ dst, float* __restrict__ deg, int E) {
  int e = blockIdx.x * blockDim.x + threadIdx.x;
  if (e < E) atomicAdd(&deg[dst[e]], 1.0f);
}

// dis[i] = rsqrt(deg[i] + 1)   (+1 = self loop; deg >= 1 always, so no zero guard)
__global__ void k_deg_finalize(float* __restrict__ deg, int n) {
  int i = blockIdx.x * blockDim.x + threadIdx.x;
  if (i < n) deg[i] = rsqrtf(deg[i] + 1.0f);
}

// ---------------- dense GEMM: T[N,128] = H[N,128] @ W[128,128] ----------------
// One block = one 16-row tile of H. 8 waves; wave w owns output columns
// [16w, 16w+16). K-loop: 32 iterations of V_WMMA_F32_16X16X4_F32.
//
// A (16x4, f32): lane m in 0..15 holds row M=m; lanes 0..15 carry K={k,k+1},
//                lanes 16..31 carry K={k+2,k+3} in VGPRs {0,1}.
// B (4x16, f32): lane m holds column N=m; same K split across lane halves.
// C/D (16x16, f32, 8 VGPRs): VGPR r, lane l -> M = r + 8*(l>>4), N = l&15.
__global__ __launch_bounds__(256) void k_gemm128_wmma(
    const float* __restrict__ H, const float* __restrict__ W,
    float* __restrict__ T, int nrows) {
  const int wave = threadIdx.x >> 5;     // 0..7 -> column tile
  const int lane = threadIdx.x & 31;
  const int half = lane >> 4;            // 0 or 1
  const int m    = lane & 15;
  const int row0 = blockIdx.x * 16;
  const int col0 = wave * 16;
  if (row0 >= nrows) return;

  // Clamp A-row for (non-multiple-of-16) safety; stores are guarded below.
  const int arow_i = min(row0 + m, nrows - 1);
  const float* __restrict__ arow = H + (size_t)arow_i * HD;
  const float* __restrict__ wcol = W + col0 + m;

  v8f acc = {};
#pragma unroll 4
  for (int k = 0; k < HD; k += 4) {
    const int ka = k + 2 * half;
    v2f a, b;
    a.x = arow[ka];
    a.y = arow[ka + 1];
    b.x = wcol[(size_t)ka * HD];
    b.y = wcol[(size_t)(ka + 1) * HD];
    // 8 args: (neg_a, A, neg_b, B, c_mod, C, reuse_a, reuse_b)
    acc = __builtin_amdgcn_wmma_f32_16x16x4_f32(
        false, a, false, b, (short)0, acc, false, false);
  }

  float* __restrict__ tcol = T + col0 + m;
  if (row0 + 16 <= nrows) {
    // Block-uniform fast path: straight-line stores, no exec juggling.
#pragma unroll
    for (int r = 0; r < 8; ++r)
      tcol[(size_t)(row0 + r + 8 * half) * HD] = acc[r];
  } else {
    // Rare partial last tile.
#pragma unroll
    for (int r = 0; r < 8; ++r) {
      const int row = row0 + r + 8 * half;
      if (row < nrows) tcol[(size_t)row * HD] = acc[r];
    }
  }
}

// ---------------- edge scatter: agg[dst] += T[src] * dis[src]*dis[dst] -------
// blockDim = 128 (one thread per feature), EDGES_PER_BLOCK edges per block.
#define EDGES_PER_BLOCK 8
__global__ __launch_bounds__(128) void k_scatter(
    const int* __restrict__ src, const int* __restrict__ dst,
    const float* __restrict__ dis, const float* __restrict__ T,
    float* __restrict__ agg, int E) {
  const int f  = threadIdx.x;
  const int e0 = blockIdx.x * EDGES_PER_BLOCK;
  const int e1 = min(e0 + EDGES_PER_BLOCK, E);
  for (int e = e0; e < e1; ++e) {
    const int s = src[e];
    const int d = dst[e];
    const float w = dis[s] * dis[d];
    atomicAdd(&agg[(size_t)d * HD + f], T[(size_t)s * HD + f] * w);
  }
}

// ---------------- epilogue: h = elu(agg + T*dis^2 + b) -----------------------
// Folds the self-loop contribution (norm = dis[i]^2) so it never hits atomics.
__global__ void k_elu_bias_selfloop(const float* __restrict__ T,
                                    const float* __restrict__ dis,
                                    const float* __restrict__ bias,
                                    float* __restrict__ agg, int n) {
  const int gid = blockIdx.x * blockDim.x + threadIdx.x;
  const int total = n * HD;
  if (gid < total) {
    const int i = gid >> 7;      // /128
    const int f = gid & (HD - 1);
    const float di = dis[i];
    const float v = agg[gid] + T[gid] * di * di + bias[f];
    agg[gid] = v > 0.0f ? v : (expf(v) - 1.0f);
  }
}

// ---------------- final GEMV: out[i] = h[i,:] @ Wl + bl ----------------------
// One wave32 per row, shuffle reduction.
__global__ __launch_bounds__(256) void k_final_gemv(
    const float* __restrict__ H, const float* __restrict__ Wl,
    const float* __restrict__ bl, float* __restrict__ out, int n) {
  const int row  = (blockIdx.x * blockDim.x + threadIdx.x) >> 5;
  const int lane = threadIdx.x & 31;
  if (row >= n) return;
  const float* __restrict__ h = H + (size_t)row * HD;
  float s = 0.0f;
#pragma unroll
  for (int f = lane; f < HD; f += 32) s += h[f] * Wl[f];
#pragma unroll
  for (int off = 16; off > 0; off >>= 1) s += __shfl_down(s, off, 32);
  if (lane == 0) out[row] = s + bl[0];
}

// ---------------- host driver ----------------

extern "C" void kernel_launch(void* const* d_in, const int* in_sizes, int n_in,
                              void* d_out, int out_size, void* d_ws, size_t ws_size,
                              hipStream_t stream) {
  const float* x  = (const float*)d_in[0];   // [N,128]
  const float* Ws = (const float*)d_in[1];   // [3,128,128]
  const float* bs = (const float*)d_in[2];   // [3,128]
  const float* Wl = (const float*)d_in[3];   // [128,1]
  const float* bl = (const float*)d_in[4];   // [1]
  const int*   ei = (const int*)d_in[5];     // [2,E]

  const int N = in_sizes[0] / HD;
  const int E = in_sizes[5] / 2;
  const int* src = ei;        // edge_index[0]
  const int* dst = ei + E;    // edge_index[1]

  // Workspace layout: dis[N] | buf0[N*128] (GEMM out t) | buf1[N*128] (agg/h)
  float* dis  = (float*)d_ws;
  float* buf0 = dis + N;
  float* buf1 = buf0 + (size_t)N * HD;

  const int TPB = 256;
  const int total = N * HD;

  // deg_inv_sqrt
  k_zero_f32<<<(N + TPB - 1) / TPB, TPB, 0, stream>>>(dis, N);
  k_deg_count<<<(E + TPB - 1) / TPB, TPB, 0, stream>>>(dst, dis, E);
  k_deg_finalize<<<(N + TPB - 1) / TPB, TPB, 0, stream>>>(dis, N);

  const int gemm_grid    = (N + 15) / 16;
  const int scatter_grid = (E + EDGES_PER_BLOCK - 1) / EDGES_PER_BLOCK;
  const int elem_grid    = (total + TPB - 1) / TPB;

  const float* hin = x;
  for (int l = 0; l < 3; ++l) {
    // t = hin @ Ws[l]
    k_gemm128_wmma<<<gemm_grid, 256, 0, stream>>>(hin, Ws + (size_t)l * HD * HD,
                                                  buf0, N);
    // agg = 0 (buf1 is free: previous h fully consumed by the GEMM above,
    // and layer 0 reads the immutable input x)
    k_zero_f32<<<elem_grid, TPB, 0, stream>>>(buf1, total);
    // agg[dst] += t[src] * dis[src]*dis[dst]
    k_scatter<<<scatter_grid, 128, 0, stream>>>(src, dst, dis, buf0, buf1, E);
    // h = elu(agg + t*dis^2 + b[l])
    k_elu_bias_selfloop<<<elem_grid, TPB, 0, stream>>>(buf0, dis, bs + l * HD,
                                                       buf1, N);
    hin = buf1;
  }

  // out = h @ Wl + bl  (one wave per row)
  const int gemv_grid = (N * 32 + 255) / 256;
  k_final_gemv<<<gemv_grid, 256, 0, stream>>>(buf1, Wl, bl, (float*)d_out, N);
}